// GcEncoder_52304111731356
// MI455X (gfx1250) — compile-verified
//
#include <hip/hip_runtime.h>
#include <hip/hip_bf16.h>
#include <math.h>

#define NUM_USERS 3000
#define NUM_ITEMS 3000
#define N_ALL     6000
#define INPUT_DIM 128
#define GCN_DIM   64

typedef __attribute__((ext_vector_type(2))) float v2f;
typedef __attribute__((ext_vector_type(8))) float v8f;

// ---------------------------------------------------------------------------
// Kernel 1: emb[i] = concat(user_X, item_X) @ wu_i, wu_i = sum_{j<=i} weights_u[j]
// One 16-row tile per block, 4 waves -> 4 column tiles of 16 (GCN_DIM=64).
// fp32 WMMA 16x16x4, K = 128 -> 32 WMMA per tile.
// ---------------------------------------------------------------------------
__global__ __launch_bounds__(128) void emb_wmma_kernel(
    const float* __restrict__ user_X,
    const float* __restrict__ item_X,
    const float* __restrict__ weights_u,
    float* __restrict__ emb)
{
    __shared__ float wu[INPUT_DIM * GCN_DIM];   // 32 KB
    const int sup     = blockIdx.y;
    const int rowBase = blockIdx.x * 16;

    // build cumulative weight matrix in LDS
    for (int t = threadIdx.x; t < INPUT_DIM * GCN_DIM; t += blockDim.x) {
        float w = weights_u[t];
        if (sup == 1) w += weights_u[INPUT_DIM * GCN_DIM + t];
        wu[t] = w;
    }
    __syncthreads();

    const int wave    = threadIdx.x >> 5;     // 0..3
    const int lane    = threadIdx.x & 31;
    const int m       = lane & 15;            // row within tile (A) / col (B,D)
    const int kh      = (lane >> 4) * 2;      // K-half select
    const int colBase = wave * 16;

    const int row = rowBase + m;
    const float* src = (row < NUM_USERS)
        ? (user_X + (size_t)row * INPUT_DIM)
        : (item_X + (size_t)(row - NUM_USERS) * INPUT_DIM);

    v8f c = {};
    for (int k = 0; k < INPUT_DIM; k += 4) {
        v2f a, b;
        a.x = src[k + kh];
        a.y = src[k + kh + 1];
        b.x = wu[(k + kh)     * GCN_DIM + colBase + m];
        b.y = wu[(k + kh + 1) * GCN_DIM + colBase + m];
        c = __builtin_amdgcn_wmma_f32_16x16x4_f32(
                false, a, false, b, (short)0, c, false, false);
    }

    float* dst = emb + (size_t)sup * N_ALL * GCN_DIM;
    const int rHalf = (lane >> 4) * 8;
    for (int v = 0; v < 8; ++v)
        dst[(size_t)(rowBase + rHalf + v) * GCN_DIM + colBase + m] = c[v];
}

// ---------------------------------------------------------------------------
// A_tilde value (only meaningful when a != 0)
// ---------------------------------------------------------------------------
__device__ __forceinline__ float atilde_val(float a, unsigned char msk,
                                            const float* __restrict__ P_symm,
                                            int r, int c)
{
    float v = a;
    if (msk) {
        unsigned rr = (unsigned)r, cc = (unsigned)c;
        unsigned idx = (cc <= rr) ? (rr * (rr + 1u) / 2u + cc)
                                  : (cc * (cc + 1u) / 2u + rr);
        float p = P_symm[idx];
        v = a / (1.0f + __expf(-p));    // a * sigmoid(p)
    }
    return v;
}

// ---------------------------------------------------------------------------
// Kernel 2: D_i[r] = sum_c A_tilde_i[r][c];  dvec = (D + 1e-7)^-0.5
// Coalesced scan of one support row per block.
// ---------------------------------------------------------------------------
__global__ __launch_bounds__(256) void rowsum_kernel(
    const float* __restrict__ support,
    const unsigned char* __restrict__ mask,
    const float* __restrict__ P_symm,
    float* __restrict__ dvec)
{
    const int r = blockIdx.x;
    const int i = blockIdx.y;
    const float* Arow        = support + (size_t)i * N_ALL * N_ALL + (size_t)r * N_ALL;
    const unsigned char* Mrow = mask + (size_t)r * N_ALL;

    float s = 0.0f;
    for (int c = threadIdx.x; c < N_ALL; c += 256) {
        float a = Arow[c];
        if (a != 0.0f) s += atilde_val(a, Mrow[c], P_symm, r, c);
    }
    __shared__ float red[256];
    red[threadIdx.x] = s;
    __syncthreads();
    for (int off = 128; off > 0; off >>= 1) {
        if (threadIdx.x < off) red[threadIdx.x] += red[threadIdx.x + off];
        __syncthreads();
    }
    if (threadIdx.x == 0)
        dvec[(size_t)i * N_ALL + r] = 1.0f / sqrtf(red[0] + 1e-7f);
}

// ---------------------------------------------------------------------------
// Kernel 3: hidden[r] = relu( sum_i d_i[r] * (A_tilde_i[r,:] @ (d_i[:] * emb_i)) )
// One wave per row: coalesced scan of A, ballot nonzeros, shfl-broadcast each
// nonzero, rank-1 update of 64-wide accumulator (2 floats per lane).
// ---------------------------------------------------------------------------
__global__ __launch_bounds__(32) void spmv_relu_kernel(
    const float* __restrict__ support,
    const unsigned char* __restrict__ mask,
    const float* __restrict__ P_symm,
    const float* __restrict__ emb,
    const float* __restrict__ dvec,
    float* __restrict__ hidden)
{
    const int r    = blockIdx.x;
    const int lane = threadIdx.x;
    const unsigned char* Mrow = mask + (size_t)r * N_ALL;

    float acc0 = 0.0f, acc1 = 0.0f;

    for (int i = 0; i < 2; ++i) {
        const float* Arow = support + (size_t)i * N_ALL * N_ALL + (size_t)r * N_ALL;
        const float* embI = emb + (size_t)i * N_ALL * GCN_DIM;
        const float* dI   = dvec + (size_t)i * N_ALL;

        float p0 = 0.0f, p1 = 0.0f;
        for (int base = 0; base < N_ALL; base += 32) {
            const int c = base + lane;
            float v = 0.0f;
            if (c < N_ALL) {
                float a = Arow[c];
                if (a != 0.0f) {
                    v = atilde_val(a, Mrow[c], P_symm, r, c);
                    v *= dI[c];                 // fold column scaling d_c
                }
            }
            unsigned long long nz = __ballot(v != 0.0f);
            while (nz) {
                const int src = __builtin_ctzll(nz);
                nz &= nz - 1;
                const float val = __shfl(v, src, 32);
                const float* e  = embI + (size_t)(base + src) * GCN_DIM + lane * 2;
                p0 = fmaf(val, e[0], p0);
                p1 = fmaf(val, e[1], p1);
            }
        }
        const float dr = dI[r];                 // fold row scaling d_r
        acc0 = fmaf(dr, p0, acc0);
        acc1 = fmaf(dr, p1, acc1);
    }

    float* h = hidden + (size_t)r * GCN_DIM + lane * 2;
    h[0] = fmaxf(acc0, 0.0f);
    h[1] = fmaxf(acc1, 0.0f);
}

// ---------------------------------------------------------------------------
// Kernel 4: out = hidden @ dense_W^T   (6000x64 @ 64x64), fp32 WMMA 16x16x4.
// B[k][n] = dense_W[n][k]; dense_W staged in LDS (16 KB).
// ---------------------------------------------------------------------------
__global__ __launch_bounds__(128) void dense_wmma_kernel(
    const float* __restrict__ hidden,
    const float* __restrict__ dense_W,
    float* __restrict__ out)
{
    __shared__ float W[GCN_DIM * GCN_DIM];
    for (int t = threadIdx.x; t < GCN_DIM * GCN_DIM; t += blockDim.x)
        W[t] = dense_W[t];
    __syncthreads();

    const int rowBase = blockIdx.x * 16;
    const int wave    = threadIdx.x >> 5;
    const int lane    = threadIdx.x & 31;
    const int m       = lane & 15;
    const int kh      = (lane >> 4) * 2;
    const int colBase = wave * 16;

    const float* h = hidden + (size_t)(rowBase + m) * GCN_DIM;

    v8f c = {};
    for (int k = 0; k < GCN_DIM; k += 4) {
        v2f a, b;
        a.x = h[k + kh];
        a.y = h[k + kh + 1];
        b.x = W[(colBase + m) * GCN_DIM + k + kh];      // dense_W[n][k]
        b.y = W[(colBase + m) * GCN_DIM + k + kh + 1];
        c = __builtin_amdgcn_wmma_f32_16x16x4_f32(
                false, a, false, b, (short)0, c, false, false);
    }

    const int rHalf = (lane >> 4) * 8;
    for (int v = 0; v < 8; ++v)
        out[(size_t)(rowBase + rHalf + v) * GCN_DIM + colBase + m] = c[v];
}

// ---------------------------------------------------------------------------
extern "C" void kernel_launch(void* const* d_in, const int* in_sizes, int n_in,
                              void* d_out, int out_size, void* d_ws, size_t ws_size,
                              hipStream_t stream) {
    const float*         user_X    = (const float*)d_in[0];
    const float*         item_X    = (const float*)d_in[1];
    const float*         support   = (const float*)d_in[2];
    const unsigned char* mask      = (const unsigned char*)d_in[3];   // bool array
    const float*         P_symm    = (const float*)d_in[4];
    const float*         weights_u = (const float*)d_in[5];
    const float*         dense_W   = (const float*)d_in[6];
    float*               out       = (float*)d_out;

    // workspace layout (floats)
    float* ws     = (float*)d_ws;
    float* emb    = ws;                         // 2*6000*64 = 768000
    float* dvec   = ws + 768000;                // 2*6000   =  12000
    float* hidden = ws + 780000;                // 6000*64  = 384000

    emb_wmma_kernel<<<dim3(N_ALL / 16, 2), 128, 0, stream>>>(
        user_X, item_X, weights_u, emb);

    rowsum_kernel<<<dim3(N_ALL, 2), 256, 0, stream>>>(
        support, mask, P_symm, dvec);

    spmv_relu_kernel<<<dim3(N_ALL), 32, 0, stream>>>(
        support, mask, P_symm, emb, dvec, hidden);

    dense_wmma_kernel<<<dim3(N_ALL / 16), 128, 0, stream>>>(
        hidden, dense_W, out);
}